// RegionFeatureExtractor_10617159155852
// MI455X (gfx1250) — compile-verified
//
#include <hip/hip_runtime.h>
#include <hip/hip_bf16.h>

typedef __attribute__((ext_vector_type(2))) float v2f;
typedef __attribute__((ext_vector_type(8))) float v8f;

#define B_   8
#define C_   256
#define HF   200
#define WF   200
#define HW   (HF * WF)
#define NBOX 100
#define SP   201   // padded integral-image side

static __device__ __forceinline__ int imin(int a, int b) { return a < b ? a : b; }
static __device__ __forceinline__ int imax(int a, int b) { return a > b ? a : b; }

// ---------------------------------------------------------------------------
// Kernel 1: 1x1-conv projection as WMMA f32 GEMM, output channels-last.
// fmt[b, hw, o] = sum_c features[b, c, hw] * proj_w[o, c] + proj_b[o]
// One wave per 16-wide hw tile; 16 accumulators cover all 256 out channels,
// so `features` is streamed from HBM exactly once.
// ---------------------------------------------------------------------------
__global__ __launch_bounds__(256) void proj_gemm_wmma(
    const float* __restrict__ features, const float* __restrict__ proj_w,
    const float* __restrict__ proj_b, float* __restrict__ fmt)
{
    const int tid  = threadIdx.x;
    const int wave = tid >> 5;
    const int lane = tid & 31;
    const int half = lane >> 4;   // 0: lanes 0-15, 1: lanes 16-31
    const int l16  = lane & 15;

    const int t   = blockIdx.x * 8 + wave;      // global wave id, 0..19999
    const int b   = t / (HW / 16);
    const int hw0 = (t % (HW / 16)) * 16;

    // B-matrix (features) base: column n = hw0 + l16, rows are channels.
    const float* fbase = features + (size_t)b * C_ * HW + hw0 + l16;

    v8f acc[16] = {};

    for (int kk = 0; kk < C_ / 4; ++kk) {
        const int k0 = kk * 4 + half * 2;       // this lane-half's K pair
        // B 4x16 tile: v0 = B[K=2h, N=l16], v1 = B[K=2h+1, N=l16]
        v2f bv;
        bv.x = fbase[(size_t)k0 * HW];
        bv.y = fbase[(size_t)(k0 + 1) * HW];

        const float* wrow = proj_w + (size_t)l16 * C_ + k0; // A row M=l16
        #pragma unroll
        for (int ot = 0; ot < 16; ++ot) {
            // A 16x4 tile: v0 = A[M=l16, K=2h], v1 = A[M=l16, K=2h+1]
            v2f av = *(const v2f*)(wrow + (size_t)ot * 16 * C_);
            acc[ot] = __builtin_amdgcn_wmma_f32_16x16x4_f32(
                false, av, false, bv, (short)0, acc[ot], false, false);
        }
    }

    // D layout: lane l16 (half h) holds D[M = half*8 + r, N = l16] in acc[r].
    // Store channels-last with bias: two b128 stores per 16x16 tile.
    float* dst = fmt + ((size_t)b * HW + hw0 + l16) * C_ + half * 8;
    #pragma unroll
    for (int ot = 0; ot < 16; ++ot) {
        const int o0 = ot * 16 + half * 8;
        float4 lo, hi;
        lo.x = acc[ot][0] + proj_b[o0 + 0];
        lo.y = acc[ot][1] + proj_b[o0 + 1];
        lo.z = acc[ot][2] + proj_b[o0 + 2];
        lo.w = acc[ot][3] + proj_b[o0 + 3];
        hi.x = acc[ot][4] + proj_b[o0 + 4];
        hi.y = acc[ot][5] + proj_b[o0 + 5];
        hi.z = acc[ot][6] + proj_b[o0 + 6];
        hi.w = acc[ot][7] + proj_b[o0 + 7];
        *(float4*)(dst + ot * 16)     = lo;
        *(float4*)(dst + ot * 16 + 4) = hi;
    }
}

// ---------------------------------------------------------------------------
// Kernel 2a: in-place cumsum along W. One block per (b, h); thread = channel.
// ---------------------------------------------------------------------------
__global__ __launch_bounds__(256) void row_cumsum(float* __restrict__ fmt)
{
    const int b = blockIdx.x / HF;
    const int h = blockIdx.x % HF;
    const int c = threadIdx.x;
    float* row = fmt + ((size_t)(b * HF + h) * WF) * C_ + c;
    float s = 0.f;
    for (int w = 0; w < WF; ++w) {
        s += row[(size_t)w * C_];
        row[(size_t)w * C_] = s;
    }
}

// ---------------------------------------------------------------------------
// Kernel 2b: cumsum along H into padded S[B,201,201,C]; also zeroes row y=0.
// ---------------------------------------------------------------------------
__global__ __launch_bounds__(256) void col_cumsum(
    const float* __restrict__ fmt, float* __restrict__ S)
{
    const int b = blockIdx.x / WF;
    const int w = blockIdx.x % WF;
    const int c = threadIdx.x;
    S[(((size_t)b * SP + 0) * SP + (w + 1)) * C_ + c] = 0.f;
    float s = 0.f;
    for (int h = 0; h < HF; ++h) {
        s += fmt[(((size_t)b * HF + h) * WF + w) * C_ + c];
        S[(((size_t)b * SP + (h + 1)) * SP + (w + 1)) * C_ + c] = s;
    }
}

// Zero the x=0 padding column of S (row y=0 was handled in col_cumsum).
__global__ __launch_bounds__(256) void pad_zero_col(float* __restrict__ S)
{
    const int b = blockIdx.x / SP;
    const int y = blockIdx.x % SP;
    S[(((size_t)b * SP + y) * SP + 0) * C_ + threadIdx.x] = 0.f;
}

// ---------------------------------------------------------------------------
// Kernel 3: per-box 7x7 adaptive average pooling via summed-area table.
// One block per (b, n); thread = channel. Corner rows are L2/L0 hot.
// ---------------------------------------------------------------------------
__global__ __launch_bounds__(256) void pool_boxes(
    const float* __restrict__ S, const int* __restrict__ boxes,
    const int* __restrict__ Hp, const int* __restrict__ Wp,
    float* __restrict__ out)
{
    const int bn = blockIdx.x;
    const int b  = bn / NBOX;
    const int c  = threadIdx.x;
    const int* bx = boxes + (size_t)bn * 4;

    const float Wi = (float)Wp[0];
    const float Hi = (float)Hp[0];
    const int x1 = bx[0], y1 = bx[1], x2 = bx[2], y2 = bx[3];

    int x1f = (int)((float)x1 / Wi * (float)WF);
    x1f = imin(imax(x1f, 0), WF - 1);
    int x2f = imax(x1f + 1, imin((int)((float)x2 / Wi * (float)WF), WF));
    int y1f = (int)((float)y1 / Hi * (float)HF);
    y1f = imin(imax(y1f, 0), HF - 1);
    int y2f = imax(y1f + 1, imin((int)((float)y2 / Hi * (float)HF), HF));
    const int hh = y2f - y1f;
    const int ww = x2f - x1f;

    const float* Sb = S + (size_t)b * SP * SP * C_ + c;
    float* ob = out + ((size_t)bn * C_ + c) * 49;

    #pragma unroll
    for (int oy = 0; oy < 7; ++oy) {
        const int ys = y1f + (oy * hh) / 7;
        const int ye = y1f + ((oy + 1) * hh + 6) / 7;
        #pragma unroll
        for (int ox = 0; ox < 7; ++ox) {
            const int xs = x1f + (ox * ww) / 7;
            const int xe = x1f + ((ox + 1) * ww + 6) / 7;
            const float v = Sb[((size_t)ye * SP + xe) * C_]
                          - Sb[((size_t)ys * SP + xe) * C_]
                          - Sb[((size_t)ye * SP + xs) * C_]
                          + Sb[((size_t)ys * SP + xs) * C_];
            const float area = (float)((ye - ys) * (xe - xs));
            ob[oy * 7 + ox] = v / area;
        }
    }
}

// Append area_ratios (second tuple output) after the pooled tensor.
__global__ void copy_tail(const float* __restrict__ ar, float* __restrict__ out)
{
    const int i = blockIdx.x * 256 + threadIdx.x;
    if (i < B_ * NBOX)
        out[(size_t)B_ * NBOX * C_ * 49 + i] = ar[i];
}

// ---------------------------------------------------------------------------
extern "C" void kernel_launch(void* const* d_in, const int* in_sizes, int n_in,
                              void* d_out, int out_size, void* d_ws, size_t ws_size,
                              hipStream_t stream)
{
    const float* features    = (const float*)d_in[0];
    const float* area_ratios = (const float*)d_in[1];
    const float* proj_w      = (const float*)d_in[2];
    const float* proj_b      = (const float*)d_in[3];
    const int*   boxes       = (const int*)d_in[4];
    const int*   Hp          = (const int*)d_in[5];
    const int*   Wp          = (const int*)d_in[6];
    float* out = (float*)d_out;

    // Workspace: fmt [B,HW,C] then padded integral image S [B,201,201,C].
    float* fmt = (float*)d_ws;
    float* S   = (float*)((char*)d_ws + (size_t)B_ * HW * C_ * sizeof(float));

    // 20000 waves (one per 16-wide hw tile), 8 waves per 256-thread block.
    proj_gemm_wmma<<<(B_ * (HW / 16)) / 8, 256, 0, stream>>>(features, proj_w, proj_b, fmt);
    row_cumsum<<<B_ * HF, 256, 0, stream>>>(fmt);
    col_cumsum<<<B_ * WF, 256, 0, stream>>>(fmt, S);
    pad_zero_col<<<B_ * SP, 256, 0, stream>>>(S);
    pool_boxes<<<B_ * NBOX, 256, 0, stream>>>(S, boxes, Hp, Wp, out);
    copy_tail<<<(B_ * NBOX + 255) / 256, 256, 0, stream>>>(area_ratios, out);
}